// MimoLinearDynamicalOperator_214748365387
// MI455X (gfx1250) — compile-verified
//
#include <hip/hip_runtime.h>
#include <hip/hip_bf16.h>

// MIMO IIR filter: FIR part via V_WMMA_F32_16X16X4_F32, AR part via exact
// block-parallel linear-recurrence decomposition (superposition).
//
// Shapes: u [B,T,CIN]=[32,16384,32], x0 [B,NA,COUT]=[32,2,32],
//         a [NA,COUT]=[2,32], bc [NB,CIN,COUT]=[3,32,32], out [B,T,COUT].

typedef float v2f __attribute__((ext_vector_type(2)));
typedef float v4f __attribute__((ext_vector_type(4)));
typedef float v8f __attribute__((ext_vector_type(8)));

constexpr int Bk   = 32;
constexpr int Tk   = 16384;
constexpr int CIN  = 32;
constexpr int COUT = 32;
constexpr int NB   = 3;
constexpr int NA   = 2;
constexpr int KTOT = NB * CIN;      // 96
constexpr int NKCH = KTOT / 4;      // 24 K-chunks of 4
constexpr int L    = 256;           // recurrence block length
constexpr int NBLK = Tk / L;        // 64

// ---------------------------------------------------------------------------
// Kernel 0: repack b_coeff into per-lane WMMA B-fragment order.
// For chunk kk, lane l (half=l>>4, lrow=l&15, k0=4kk+2*half):
//   bp[kk][l][0]=B[k0][lrow]     bp[kk][l][1]=B[k0+1][lrow]
//   bp[kk][l][2]=B[k0][lrow+16]  bp[kk][l][3]=B[k0+1][lrow+16]
// so the hot loop does a single b128 load per chunk.
// ---------------------------------------------------------------------------
__global__ void bpack_kernel(const float* __restrict__ bc,
                             float* __restrict__ bp) {
  const int idx = blockIdx.x * blockDim.x + threadIdx.x;  // 0..3071
  if (idx >= NKCH * 32 * 4) return;
  const int j    = idx & 3;
  const int lane = (idx >> 2) & 31;
  const int kk   = idx >> 7;
  const int half = lane >> 4, lrow = lane & 15;
  const int k0   = kk * 4 + half * 2;
  const int krow = k0 + (j & 1);
  const int n    = lrow + ((j >> 1) << 4);
  bp[idx] = bc[(size_t)krow * COUT + n];
}

// ---------------------------------------------------------------------------
// Kernel 1: FIR  f[b,t,o] = sum_{d<NB} sum_i u[b,t-d,i] * bc[d,i,o]
// GEMM: M = time (16 rows/tile), N = COUT (2x16), K = 96 in chunks of 4.
// A 16x4 f32 frag (ISA 7.12.2): lanes 0-15 row M=lane  {K=4kk,4kk+1},
//                               lanes16-31 row M=lane-16{K=4kk+2,4kk+3}.
// k0 even => both A elements share d and are channel-adjacent -> one b64 load.
// ---------------------------------------------------------------------------
template <bool GUARD>
__device__ __forceinline__ void fir_tile(const float* __restrict__ ub,
                                         const float* __restrict__ bp,
                                         int trow, int half, int lane,
                                         v8f& acc0, v8f& acc1) {
  const float* arow = ub + (size_t)trow * CIN;  // A row (time = trow)
#pragma unroll
  for (int kk = 0; kk < NKCH; ++kk) {
    const int k0 = kk * 4 + half * 2;           // lane supplies K=k0,k0+1
    const int d0 = k0 >> 5, i0 = k0 & 31;       // same d for both elements

    v2f a;
    if (GUARD) {
      // first tile only: branch-free masked load (keeps EXEC uniform)
      const float msk   = (trow >= d0) ? 1.0f : 0.0f;
      const int   tsafe = (trow >= d0) ? (trow - d0) : 0;
      a = *(const v2f*)(ub + (size_t)tsafe * CIN + i0);
      a.x *= msk; a.y *= msk;
    } else {
      a = *(const v2f*)(arow + (i0 - d0 * CIN));
    }

    const v4f bv = *(const v4f*)(bp + (((size_t)kk * 32 + lane) << 2));
    const v2f b0 = {bv.x, bv.y};   // N-tile 0 (cols 0-15)
    const v2f b1 = {bv.z, bv.w};   // N-tile 1 (cols 16-31)

    acc0 = __builtin_amdgcn_wmma_f32_16x16x4_f32(false, a, false, b0,
                                                 (short)0, acc0, false, false);
    acc1 = __builtin_amdgcn_wmma_f32_16x16x4_f32(false, a, false, b1,
                                                 (short)0, acc1, false, false);
  }
}

__global__ __launch_bounds__(128) void fir_wmma_kernel(
    const float* __restrict__ u, const float* __restrict__ bp,
    float* __restrict__ f) {
  const int wave = threadIdx.x >> 5;
  const int lane = threadIdx.x & 31;
  const int tile = blockIdx.x * 4 + wave;      // 0 .. B*(T/16)-1
  const int b    = tile >> 10;                 // tile / (T/16)
  const int t0   = (tile & 1023) << 4;

  const int half = lane >> 4;
  const int lrow = lane & 15;
  const int trow = t0 + lrow;

  const float* ub = u + (size_t)b * Tk * CIN;

  v8f acc0 = {};   // N = 0..15
  v8f acc1 = {};   // N = 16..31

  if (t0 >= NB - 1) {          // wave-uniform branch: 1023/1024 tiles
    fir_tile<false>(ub, bp, trow, half, lane, acc0, acc1);
  } else {                     // first tile of each batch: t-d may be < 0
    fir_tile<true>(ub, bp, trow, half, lane, acc0, acc1);
  }

  // C/D layout: VGPR r -> M = r + 8*half, N = lrow (+16 for tile 1)
#pragma unroll
  for (int r = 0; r < 8; ++r) {
    const int m = r + half * 8;
    const size_t base = ((size_t)b * Tk + (t0 + m)) * COUT;
    f[base + lrow]      = acc0[r];
    f[base + lrow + 16] = acc1[r];
  }
}

// ---------------------------------------------------------------------------
// Kernel 2: homogeneous responses per channel.
// h1_t: recurrence with (x_{-1},x_{-2})=(1,0), f=0;  h2_t: (0,1).
// ---------------------------------------------------------------------------
__global__ void hresp_kernel(const float* __restrict__ a_coeff,
                             float* __restrict__ h1, float* __restrict__ h2) {
  const int o = threadIdx.x;
  if (o >= COUT) return;
  const float a1 = a_coeff[o], a2 = a_coeff[COUT + o];
  float p1 = 1.f, p2 = 0.f;    // h1 state
  float q1 = 0.f, q2 = 1.f;    // h2 state
  for (int t = 0; t < L; ++t) {
    const float hp = a1 * p1 + a2 * p2;
    const float hq = a1 * q1 + a2 * q2;
    h1[t * COUT + o] = hp;
    h2[t * COUT + o] = hq;
    p2 = p1; p1 = hp;
    q2 = q1; q1 = hq;
  }
}

// ---------------------------------------------------------------------------
// Kernel 3: per-block zero-init scan; store only the block-end partials
// (particular solution at tau = L-1 and L-2). One thread per (b,k,o).
// ---------------------------------------------------------------------------
__global__ __launch_bounds__(256) void block_partial_kernel(
    const float* __restrict__ f, const float* __restrict__ a_coeff,
    float* __restrict__ pend1, float* __restrict__ pend2) {
  const int g    = blockIdx.x * 256 + threadIdx.x;
  const int o    = g & 31;
  const int pair = g >> 5;
  const int k    = pair & (NBLK - 1);
  const int b    = pair >> 6;
  const float a1 = a_coeff[o], a2 = a_coeff[COUT + o];
  const size_t base = ((size_t)b * Tk + (size_t)k * L) * COUT + o;
  float y1 = 0.f, y2 = 0.f;
#pragma unroll 4
  for (int tau = 0; tau < L; ++tau) {
    const float y = f[base + (size_t)tau * COUT] + a1 * y1 + a2 * y2;
    y2 = y1; y1 = y;
  }
  const size_t pidx = ((size_t)k * Bk + b) * COUT + o;
  pend1[pidx] = y1;   // tau = L-1
  pend2[pidx] = y2;   // tau = L-2
}

// ---------------------------------------------------------------------------
// Kernel 4: chain block boundaries. x_{kL-1}, x_{kL-2} per (b,o):
//   s_{k+1} = pend_k + H * s_k  (2x2 per channel). 64 serial steps only.
// ---------------------------------------------------------------------------
__global__ void boundary_kernel(
    const float* __restrict__ x0,
    const float* __restrict__ h1, const float* __restrict__ h2,
    const float* __restrict__ pend1, const float* __restrict__ pend2,
    float* __restrict__ init1, float* __restrict__ init2) {
  const int g = blockIdx.x * blockDim.x + threadIdx.x;   // 0..1023
  const int o = g & 31;
  const int b = g >> 5;
  const float hL1_1 = h1[(L - 1) * COUT + o], hL1_2 = h2[(L - 1) * COUT + o];
  const float hL2_1 = h1[(L - 2) * COUT + o], hL2_2 = h2[(L - 2) * COUT + o];
  float s1 = x0[((size_t)b * NA + 1) * COUT + o];   // x_{-1}
  float s2 = x0[((size_t)b * NA + 0) * COUT + o];   // x_{-2}
  for (int k = 0; k < NBLK; ++k) {
    const size_t pidx = ((size_t)k * Bk + b) * COUT + o;
    init1[pidx] = s1;
    init2[pidx] = s2;
    const float n1 = pend1[pidx] + hL1_1 * s1 + hL1_2 * s2;  // x_{(k+1)L-1}
    const float n2 = pend2[pidx] + hL2_1 * s1 + hL2_2 * s2;  // x_{(k+1)L-2}
    s1 = n1; s2 = n2;
  }
}

// ---------------------------------------------------------------------------
// Kernel 5: final solve — re-run each block with its true initial state,
// writing the output in place over f (which lives in d_out).
// ---------------------------------------------------------------------------
__global__ __launch_bounds__(256) void block_solve_kernel(
    float* __restrict__ f, const float* __restrict__ a_coeff,
    const float* __restrict__ init1, const float* __restrict__ init2) {
  const int g    = blockIdx.x * 256 + threadIdx.x;
  const int o    = g & 31;
  const int pair = g >> 5;
  const int k    = pair & (NBLK - 1);
  const int b    = pair >> 6;
  const float a1 = a_coeff[o], a2 = a_coeff[COUT + o];
  const size_t pidx = ((size_t)k * Bk + b) * COUT + o;
  float y1 = init1[pidx], y2 = init2[pidx];
  const size_t base = ((size_t)b * Tk + (size_t)k * L) * COUT + o;
#pragma unroll 4
  for (int tau = 0; tau < L; ++tau) {
    const size_t idx = base + (size_t)tau * COUT;
    const float y = f[idx] + a1 * y1 + a2 * y2;
    f[idx] = y;
    y2 = y1; y1 = y;
  }
}

// ---------------------------------------------------------------------------
extern "C" void kernel_launch(void* const* d_in, const int* in_sizes, int n_in,
                              void* d_out, int out_size, void* d_ws,
                              size_t ws_size, hipStream_t stream) {
  const float* u  = (const float*)d_in[0];   // [B,T,CIN]
  const float* x0 = (const float*)d_in[1];   // [B,NA,COUT]
  const float* ac = (const float*)d_in[2];   // [NA,COUT]
  const float* bc = (const float*)d_in[3];   // [NB,CIN,COUT]
  float* out = (float*)d_out;                // [B,T,COUT] (also FIR scratch)

  // Workspace carve-up (floats): ~1.08 MB total. bp first (16B aligned).
  float* ws    = (float*)d_ws;
  float* bp    = ws;                               // [NKCH][32][4] = 3072
  float* h1    = bp + NKCH * 32 * 4;               // [L][COUT]
  float* h2    = h1 + L * COUT;                    // [L][COUT]
  float* pend1 = h2 + L * COUT;                    // [NBLK][B][COUT]
  float* pend2 = pend1 + NBLK * Bk * COUT;
  float* init1 = pend2 + NBLK * Bk * COUT;
  float* init2 = init1 + NBLK * Bk * COUT;
  (void)ws_size; (void)in_sizes; (void)n_in; (void)out_size;

  // 0) Repack B matrix into per-lane fragment order (12 KB, L2-resident).
  bpack_kernel<<<(NKCH * 32 * 4 + 255) / 256, 256, 0, stream>>>(bc, bp);
  // 1) FIR via WMMA into d_out.  B*(T/16)=32768 wave-tiles, 4 waves/block.
  fir_wmma_kernel<<<(Bk * (Tk / 16)) / 4, 128, 0, stream>>>(u, bp, out);
  // 2) Homogeneous responses (32 channels, one wave).
  hresp_kernel<<<1, 32, 0, stream>>>(ac, h1, h2);
  // 3) Per-block zero-init partials. NBLK*B*COUT = 65536 threads.
  block_partial_kernel<<<(NBLK * Bk * COUT) / 256, 256, 0, stream>>>(
      out, ac, pend1, pend2);
  // 4) Boundary chain: 1024 threads, 64 serial 2x2 steps.
  boundary_kernel<<<4, 256, 0, stream>>>(x0, h1, h2, pend1, pend2, init1, init2);
  // 5) Final in-place solve with true initial states.
  block_solve_kernel<<<(NBLK * Bk * COUT) / 256, 256, 0, stream>>>(
      out, ac, init1, init2);
}